// lossFunction_71219147702889
// MI455X (gfx1250) — compile-verified
//
#include <hip/hip_runtime.h>

// YOLO-style loss, S=1024, DEPTH=30. Bandwidth-bound streaming reduction:
// 252 MB read @ 23.3 TB/s => ~10.8us floor. 2 cells = 240B = 15 x float4
// (16B aligned) so each thread consumes a 2-cell group with b128 loads.
// Wave reduction done with V_WMMA_F32_16X16X4_F32 (A = lane partials,
// B = ones) + one shfl_xor(16).

#define W_CLS  1.5f   // 1 + LAMBDA_NOOBJ
#define W_BOX  5.0f   // LAMBDA_COORD

typedef float v2f __attribute__((ext_vector_type(2)));
typedef float v8f __attribute__((ext_vector_type(8)));

// Full 32-lane sum of `partial`, result broadcast to all lanes.
// A (16x4 f32): lane L (0-15): VGPR0=A[L][0], VGPR1=A[L][1];
//               lane L (16-31): VGPR0=A[L-16][2], VGPR1=A[L-16][3].
// With a = {partial, 0} and B = all ones:
//   D[m][n] = partial(m) + partial(m+16)  (all columns equal)
// D layout: VGPR r -> row r (lanes 0-15) / row r+8 (lanes 16-31), so
// sum of d[0..7] = half-wave sum; shfl_xor(16) completes the reduction.
__device__ __forceinline__ float wave_reduce_wmma(float partial) {
    v2f a; a.x = partial; a.y = 0.0f;
    v2f b; b.x = 1.0f;    b.y = 1.0f;
    v8f c = {};
    v8f d = __builtin_amdgcn_wmma_f32_16x16x4_f32(
        /*neg_a=*/false, a, /*neg_b=*/false, b,
        /*c_mod=*/(short)0, c, /*reuse_a=*/false, /*reuse_b=*/false);
    float t = ((d[0] + d[1]) + (d[2] + d[3])) + ((d[4] + d[5]) + (d[6] + d[7]));
    t += __shfl_xor(t, 16, 32);
    return t;
}

// Faithful to the reference's nonstandard corner/sign conventions.
__device__ __forceinline__ float iou_term(float px, float py, float pw, float ph,
                                          float gx, float gy, float gw, float gh) {
    float px1 = px - pw * 0.5f, px2 = px + pw * 0.5f;
    float py1 = py - ph * 0.5f, py2 = py + ph * 0.5f;
    float gx1 = gx - gw * 0.5f, gx2 = gx + gw * 0.5f;
    float gy1 = gy - gh * 0.5f, gy2 = gy + gh * 0.5f;
    float tlx = fmaxf(gx1, px1);
    float tly = fminf(gy1, py1);
    float brx = fmaxf(gx2, px2);
    float bry = fminf(gy2, py2);
    float inter = (brx - tlx) * (tly - bry);
    float uni   = pw * ph + gw * gh - inter;
    return inter / uni;
}

__global__ void yolo_loss_partial(const float* __restrict__ y_pred,
                                  const float* __restrict__ y,
                                  float* __restrict__ partials,
                                  int num_pairs) {
    const float4* __restrict__ P = (const float4*)y_pred;
    const float4* __restrict__ G = (const float4*)y;

    float acc = 0.0f;
    int stride = (int)(gridDim.x * blockDim.x);
    for (int u = (int)(blockIdx.x * blockDim.x + threadIdx.x); u < num_pairs; u += stride) {
        const float4* p = P + (size_t)u * 15;
        const float4* g = G + (size_t)u * 15;

        float cls = 0.0f, box = 0.0f;
        float p0x, p0y, p0w, p0h, g0x, g0y, g0w, g0h;
        float p1x, p1y, p1w, p1h, g1x, g1y, g1w, g1h;
        float4 pv, gv;
        float dx, dy, dz, dw;

        // Quad q covers flat channels [4q, 4q+3] of the 60-channel 2-cell group:
        // cell0 = ch 0..29, cell1 = ch 30..59; per cell: 0..19 class, 20..29 box,
        // box x,y,w,h at channels 20..23.
#define LOADQ(q) pv = p[q]; gv = g[q]; \
                 dx = gv.x - pv.x; dy = gv.y - pv.y; \
                 dz = gv.z - pv.z; dw = gv.w - pv.w;
#define SUM4 (dx * dx + dy * dy + dz * dz + dw * dw)

        LOADQ(0)  cls += SUM4;                       // cell0 ch0-3   class
        LOADQ(1)  cls += SUM4;                       // cell0 ch4-7   class
        LOADQ(2)  cls += SUM4;                       // cell0 ch8-11  class
        LOADQ(3)  cls += SUM4;                       // cell0 ch12-15 class
        LOADQ(4)  cls += SUM4;                       // cell0 ch16-19 class
        LOADQ(5)  box += SUM4;                       // cell0 ch20-23 box (x,y,w,h)
                  p0x = pv.x; p0y = pv.y; p0w = pv.z; p0h = pv.w;
                  g0x = gv.x; g0y = gv.y; g0w = gv.z; g0h = gv.w;
        LOADQ(6)  box += SUM4;                       // cell0 ch24-27 box
        LOADQ(7)  box += dx * dx + dy * dy;          // cell0 ch28-29 box
                  cls += dz * dz + dw * dw;          // cell1 ch0-1   class
        LOADQ(8)  cls += SUM4;                       // cell1 ch2-5   class
        LOADQ(9)  cls += SUM4;                       // cell1 ch6-9   class
        LOADQ(10) cls += SUM4;                       // cell1 ch10-13 class
        LOADQ(11) cls += SUM4;                       // cell1 ch14-17 class
        LOADQ(12) cls += dx * dx + dy * dy;          // cell1 ch18-19 class
                  box += dz * dz + dw * dw;          // cell1 ch20-21 box (x,y)
                  p1x = pv.z; p1y = pv.w; g1x = gv.z; g1y = gv.w;
        LOADQ(13) box += SUM4;                       // cell1 ch22-25 box (w,h,...)
                  p1w = pv.x; p1h = pv.y; g1w = gv.x; g1h = gv.y;
        LOADQ(14) box += SUM4;                       // cell1 ch26-29 box

#undef LOADQ
#undef SUM4

        acc += W_CLS * cls + W_BOX * box
             + iou_term(p0x, p0y, p0w, p0h, g0x, g0y, g0w, g0h)
             + iou_term(p1x, p1y, p1w, p1h, g1x, g1y, g1w, g1h);
    }

    // Uniform control flow from here: EXEC all-1s, full waves -> WMMA legal.
    float wsum = wave_reduce_wmma(acc);

    __shared__ float sdata[8];
    int lane = (int)(threadIdx.x & 31u);
    int wid  = (int)(threadIdx.x >> 5);
    if (lane == 0) sdata[wid] = wsum;
    __syncthreads();
    if (threadIdx.x == 0) {
        float s = 0.0f;
#pragma unroll
        for (int i = 0; i < 8; ++i) s += sdata[i];
        partials[blockIdx.x] = s;
    }
}

__global__ void reduce_partials(const float* __restrict__ partials,
                                float* __restrict__ out, int n) {
    float acc = 0.0f;
    for (int i = (int)threadIdx.x; i < n; i += (int)blockDim.x) acc += partials[i];

    float wsum = wave_reduce_wmma(acc);

    __shared__ float sdata[8];
    int lane = (int)(threadIdx.x & 31u);
    int wid  = (int)(threadIdx.x >> 5);
    if (lane == 0) sdata[wid] = wsum;
    __syncthreads();
    if (threadIdx.x == 0) {
        float s = 0.0f;
#pragma unroll
        for (int i = 0; i < 8; ++i) s += sdata[i];
        out[0] = s;
    }
}

extern "C" void kernel_launch(void* const* d_in, const int* in_sizes, int n_in,
                              void* d_out, int out_size, void* d_ws, size_t ws_size,
                              hipStream_t stream) {
    const float* y_pred = (const float*)d_in[0];
    const float* y      = (const float*)d_in[1];
    float* out      = (float*)d_out;
    float* partials = (float*)d_ws;   // NB floats (4 KB)

    int n_elems   = in_sizes[0];      // S*S*DEPTH = 31,457,280
    int num_pairs = n_elems / 60;     // 2-cell groups: 524,288

    const int NB = 1024;              // fixed -> exactly NB partials, deterministic
    const int NT = 256;               // 8 wave32s per block

    yolo_loss_partial<<<NB, NT, 0, stream>>>(y_pred, y, partials, num_pairs);
    reduce_partials<<<1, NT, 0, stream>>>(partials, out, NB);
}